// GuidedFilterLayer_86028194939333
// MI455X (gfx1250) — compile-verified
//
#include <hip/hip_runtime.h>
#include <hip/hip_bf16.h>

typedef float v2f __attribute__((ext_vector_type(2)));
typedef float v8f __attribute__((ext_vector_type(8)));

#define IMG_H 512
#define IMG_W 512
#define N_IMG 16
#define RADIUS 7
#define W15 (1.0f / 15.0f)

// ---------------- Pass 1: per-block partial channel sums ----------------
__global__ void gf_partial_sums(const float* __restrict__ x,
                                float* __restrict__ partials, int numPix) {
    __shared__ float shr[256], shg[256], shb[256];
    const int tid = threadIdx.x;
    float sr = 0.f, sg = 0.f, sb = 0.f;
    for (int p = blockIdx.x * blockDim.x + tid; p < numPix;
         p += gridDim.x * blockDim.x) {
        const float* px = x + (size_t)p * 3;
        sr += px[0]; sg += px[1]; sb += px[2];
    }
    shr[tid] = sr; shg[tid] = sg; shb[tid] = sb;
    __syncthreads();
    for (int s = 128; s > 0; s >>= 1) {
        if (tid < s) {
            shr[tid] += shr[tid + s];
            shg[tid] += shg[tid + s];
            shb[tid] += shb[tid + s];
        }
        __syncthreads();
    }
    if (tid == 0) {
        partials[blockIdx.x * 3 + 0] = shr[0];
        partials[blockIdx.x * 3 + 1] = shg[0];
        partials[blockIdx.x * 3 + 2] = shb[0];
    }
}

// ---------------- Pass 2: finalize scalar c = mean(inputs) - mean(guidance) ----
__global__ void gf_finalize(const float* __restrict__ partials,
                            float* __restrict__ cOut, int nblk, int numPix) {
    __shared__ float shr[256], shg[256], shb[256];
    const int tid = threadIdx.x;
    float sr = 0.f, sg = 0.f, sb = 0.f;
    for (int i = tid; i < nblk; i += 256) {
        sr += partials[i * 3 + 0];
        sg += partials[i * 3 + 1];
        sb += partials[i * 3 + 2];
    }
    shr[tid] = sr; shg[tid] = sg; shb[tid] = sb;
    __syncthreads();
    for (int s = 128; s > 0; s >>= 1) {
        if (tid < s) {
            shr[tid] += shr[tid + s];
            shg[tid] += shg[tid + s];
            shb[tid] += shb[tid + s];
        }
        __syncthreads();
    }
    if (tid == 0) {
        const float inv = 1.0f / (float)numPix;
        // channel means of inputs = (x+1)*0.5
        float mr = fmaf(shr[0] * inv, 0.5f, 0.5f);
        float mg = fmaf(shg[0] * inv, 0.5f, 0.5f);
        float mb = fmaf(shb[0] * inv, 0.5f, 0.5f);
        float mean_in = (mr + mg + mb) * (1.0f / 3.0f);
        float mean_gd = 0.2989f * mr + 0.5870f * mg + 0.1140f * mb;
        cOut[0] = mean_in - mean_gd;
    }
}

// ---------------- Pass 3: fused guidance + 15x15 box blur (WMMA) + mix ------
// Block: 256 threads (8 waves) -> 32 rows x 64 cols of output.
// Each wave: one 16x16 tile. Blur = two banded GEMMs on v_wmma_f32_16x16x4_f32.
#define GS_ROWS 48   // 32 + 2*7 (+2 pad rows consumed by K padding)
#define GS_COLS 80   // 64 + 2*7 (+2 pad cols consumed by K padding)
#define GS_STRIDE 81
#define HS_STRIDE 17

__global__ void __launch_bounds__(256)
gf_main(const float* __restrict__ x, const float* __restrict__ cPtr,
        float* __restrict__ out) {
    __shared__ float Gs[GS_ROWS * GS_STRIDE];      // adjusted guidance w/ halo
    __shared__ float Hs[8 * 32 * HS_STRIDE];       // per-wave H-blur scratch

    const int tid  = threadIdx.x;
    const int img  = blockIdx.z;
    const int gy0  = blockIdx.y * 32;
    const int gx0  = blockIdx.x * 64;
    const float cv = cPtr[0];

    // ---- Phase 1: cooperative guidance fill (zero-padded, +c) ----
    for (int i = tid; i < GS_ROWS * GS_COLS; i += 256) {
        const int ry = i / GS_COLS;
        const int rx = i - ry * GS_COLS;
        const int iy = gy0 + ry - RADIUS;
        const int ix = gx0 + rx - RADIUS;
        float g = 0.0f;
        if (iy >= 0 && iy < IMG_H && ix >= 0 && ix < IMG_W) {
            const float* p =
                x + (((size_t)img * IMG_H + iy) * IMG_W + ix) * 3;
            float r = fmaf(p[0], 0.5f, 0.5f);
            float gg = fmaf(p[1], 0.5f, 0.5f);
            float b = fmaf(p[2], 0.5f, 0.5f);
            g = fmaf(0.2989f, r, fmaf(0.5870f, gg, fmaf(0.1140f, b, cv)));
        }
        Gs[ry * GS_STRIDE + rx] = g;
    }
    __syncthreads();

    // ---- Phase 2: per-wave separable blur via WMMA f32 16x16x4 ----
    const int wid  = tid >> 5;        // wave id 0..7
    const int lane = tid & 31;
    const int ty   = wid >> 2;        // 0..1 (16-row tile)
    const int tx   = wid & 3;         // 0..3 (16-col tile)
    const int m    = lane & 15;       // M (A) / N (B,D) index
    const int h    = lane >> 4;       // lane half
    float* hsw = &Hs[wid * 32 * HS_STRIDE];

    // Horizontal pass: H(32x16) = G(32x32-band) x Bh, done as 2 row groups
    #pragma unroll
    for (int rg = 0; rg < 2; ++rg) {
        v8f acc = {};
        const int ryB = ty * 16 + rg * 16 + m;   // Gs row for this lane's M
        #pragma unroll
        for (int kc = 0; kc < 8; ++kc) {
            const int K0 = 4 * kc + 2 * h;       // K index for VGPR j=0
            v2f a, b;
            a.x = Gs[ryB * GS_STRIDE + tx * 16 + K0];
            a.y = Gs[ryB * GS_STRIDE + tx * 16 + K0 + 1];
            b.x = (K0 >= m && K0 <= m + 14) ? W15 : 0.0f;
            b.y = (K0 + 1 >= m && K0 + 1 <= m + 14) ? W15 : 0.0f;
            acc = __builtin_amdgcn_wmma_f32_16x16x4_f32(
                false, a, false, b, (short)0, acc, false, false);
        }
        // D layout: VGPR v -> row v (lanes 0-15) / v+8 (lanes 16-31), col m
        #pragma unroll
        for (int v = 0; v < 8; ++v)
            hsw[(rg * 16 + v + h * 8) * HS_STRIDE + m] = acc[v];
    }

    // Vertical pass: S(16x16) = Bv(16x32-band) x H(32x16)
    v8f acc2 = {};
    #pragma unroll
    for (int kc = 0; kc < 8; ++kc) {
        const int K0 = 4 * kc + 2 * h;
        v2f a, b;
        a.x = (K0 >= m && K0 <= m + 14) ? W15 : 0.0f;
        a.y = (K0 + 1 >= m && K0 + 1 <= m + 14) ? W15 : 0.0f;
        b.x = hsw[K0 * HS_STRIDE + m];
        b.y = hsw[(K0 + 1) * HS_STRIDE + m];
        acc2 = __builtin_amdgcn_wmma_f32_16x16x4_f32(
            false, a, false, b, (short)0, acc2, false, false);
    }

    // ---- Phase 3: mix and store: out = 0.99*x + 0.02*s - 0.01 ----
    const int oy0 = gy0 + ty * 16;
    const int ox  = gx0 + tx * 16 + m;
    #pragma unroll
    for (int v = 0; v < 8; ++v) {
        const int oy = oy0 + v + h * 8;
        const size_t base = (((size_t)img * IMG_H + oy) * IMG_W + ox) * 3;
        const float s = acc2[v];
        const float add = fmaf(0.02f, s, -0.01f);
        out[base + 0] = fmaf(0.99f, x[base + 0], add);
        out[base + 1] = fmaf(0.99f, x[base + 1], add);
        out[base + 2] = fmaf(0.99f, x[base + 2], add);
    }
}

// ---------------- Launch ----------------
extern "C" void kernel_launch(void* const* d_in, const int* in_sizes, int n_in,
                              void* d_out, int out_size, void* d_ws, size_t ws_size,
                              hipStream_t stream) {
    const float* x = (const float*)d_in[0];
    float* out = (float*)d_out;
    float* ws = (float*)d_ws;

    const int numPix = N_IMG * IMG_H * IMG_W;
    const int NBLK = 512;
    float* partials = ws;            // NBLK*3 floats
    float* cScalar  = ws + NBLK * 3; // 1 float

    gf_partial_sums<<<NBLK, 256, 0, stream>>>(x, partials, numPix);
    gf_finalize<<<1, 256, 0, stream>>>(partials, cScalar, NBLK, numPix);

    dim3 grid(IMG_W / 64, IMG_H / 32, N_IMG);
    gf_main<<<grid, 256, 0, stream>>>(x, cScalar, out);
}